// MultiHeadAttention_3272765080134
// MI455X (gfx1250) — compile-verified
//
#include <hip/hip_runtime.h>
#include <math.h>

typedef float v2f __attribute__((ext_vector_type(2)));
typedef float v8f __attribute__((ext_vector_type(8)));
typedef float f4  __attribute__((ext_vector_type(4)));

#define NEG_BIG (-3.0e38f)

// LDS geometry (all strides chosen so every WMMA operand is one aligned,
// bank-conflict-free ds_load_b64):
//   row stride 20 floats: 20*r mod 64 hits 16 disjoint 4-bank spans
//   pair stride 160 floats: 160 mod 64 == 32 -> lane halves hit disjoint banks
#define R_STRIDE 20
#define B_STRIDE 160

// One CDNA5 async DMA: 16 bytes global -> LDS per lane, tracked by ASYNCcnt.
__device__ __forceinline__ void asyncLoad16(unsigned ldsOff, const float* g)
{
    asm volatile("global_load_async_to_lds_b128 %0, %1, off"
                 :: "v"(ldsOff), "v"(g)
                 : "memory");
}

// ---------------------------------------------------------------------------
// Generic 64x64-tile fp32 WMMA GEMM:  C[z] = A[z] * B[z] (optionally B^T),
// double-buffered async (TDM-path) staging, optional causal-mask+scale or
// bias+residual epilogues.  128 threads = 4 waves; wave w owns rows
// [16w,16w+16).  All dims are multiples of 64/16 -> no bounds checks.
// ---------------------------------------------------------------------------
template<bool TRANSB, bool CAUSAL, bool BIASRES>
__global__ __launch_bounds__(128)
void gemm_wmma(const float* __restrict__ A, const float* __restrict__ B,
               float* __restrict__ C, int K,
               int lda, int ldb, int ldc,
               long sA, long sB, long sC, int cColPerZ,
               float scale,
               const float* __restrict__ bias,
               const float* __restrict__ resid, int ldres)
{
    __shared__ __align__(16) float Alds[2][64 * R_STRIDE];   // [buf][row][k] (async)
    __shared__ __align__(16) float BldsT[2][64 * R_STRIDE];  // [buf][n][k]   (async, TRANSB)
    __shared__ __align__(16) float Blds[8 * B_STRIDE];       // [kpair][n][2] (manual, !TRANSB)

    const int tid  = threadIdx.x;
    const int wave = tid >> 5;
    const int lane = tid & 31;
    const int lh   = (lane >> 4) & 1;   // lane half (wave32)
    const int l16  = lane & 15;

    const int mBase = blockIdx.y << 6;
    const int nBase = blockIdx.x << 6;
    const int z     = blockIdx.z;

    const float* Ah = A + (size_t)z * sA;
    const float* Bh = B + (size_t)z * sB;
    float*       Ch = C + (size_t)z * sC + (size_t)z * cColPerZ;

    const int rowM = mBase + (wave << 4) + (lh << 3);  // first of 8 C-rows this lane owns

    if (CAUSAL && (nBase > mBase + 63)) {
        // tile entirely above the diagonal: emit sentinel, skip all math
        for (int j = 0; j < 4; ++j) {
            int n = nBase + (j << 4) + l16;
            for (int r = 0; r < 8; ++r)
                Ch[(size_t)(rowM + r) * ldc + n] = NEG_BIG;
        }
        return;
    }

    const unsigned aBase[2] = { (unsigned)(uintptr_t)&Alds[0][0],
                                (unsigned)(uintptr_t)&Alds[1][0] };
    const unsigned bBase[2] = { (unsigned)(uintptr_t)&BldsT[0][0],
                                (unsigned)(uintptr_t)&BldsT[1][0] };

    // ---- async issue: A 64x16 slab (2 wave-ops), and B^T slab when TRANSB ----
    auto issueA = [&](int k0, int buf) {
        #pragma unroll
        for (int i = 0; i < 2; ++i) {
            int c = tid + (i << 7);
            int r = c >> 2, q4 = (c & 3) << 2;
            asyncLoad16(aBase[buf] + (unsigned)((r * R_STRIDE + q4) << 2),
                        Ah + (size_t)(mBase + r) * lda + k0 + q4);
        }
    };
    auto issueBT = [&](int k0, int buf) {
        #pragma unroll
        for (int i = 0; i < 2; ++i) {
            int c = tid + (i << 7);
            int n = c >> 2, q4 = (c & 3) << 2;
            asyncLoad16(bBase[buf] + (unsigned)((n * R_STRIDE + q4) << 2),
                        Bh + (size_t)(nBase + n) * ldb + k0 + q4);
        }
    };

    // ---- manual (pipelined) staging for non-transposed B: K-pair interleave ----
    f4 breg[2];
    auto loadB = [&](int k0) {
        #pragma unroll
        for (int i = 0; i < 2; ++i) {
            int c = tid + (i << 7);
            int r = c >> 4, c4 = (c & 15) << 2;
            breg[i] = *(const f4*)(Bh + (size_t)(k0 + r) * ldb + nBase + c4);
        }
    };
    auto storeB = [&]() {
        #pragma unroll
        for (int i = 0; i < 2; ++i) {
            int c = tid + (i << 7);
            int r = c >> 4, c4 = (c & 15) << 2;
            float* dst = &Blds[(r >> 1) * B_STRIDE + (r & 1)];
            dst[2 * (c4 + 0)] = breg[i].x;
            dst[2 * (c4 + 1)] = breg[i].y;
            dst[2 * (c4 + 2)] = breg[i].z;
            dst[2 * (c4 + 3)] = breg[i].w;
        }
    };

    v8f acc[4] = {};
    const int aRowOff = ((wave << 4) + l16) * R_STRIDE + (lh << 1);

    // prologue: stage slab 0
    issueA(0, 0);
    if constexpr (TRANSB) issueBT(0, 0); else loadB(0);

    int buf = 0;
    for (int k0 = 0; k0 < K; k0 += 16) {
        const bool hasNext = (k0 + 16) < K;

        // prefetch slab k0+16 into the other buffer (runs during compute)
        if (hasNext) {
            issueA(k0 + 16, buf ^ 1);
            if constexpr (TRANSB) issueBT(k0 + 16, buf ^ 1);
        }
        if constexpr (!TRANSB) {
            storeB();                       // compiler waits LOADcnt for bregs
            if (hasNext) loadB(k0 + 16);    // overlaps with compute below
        }

        // oldest slab's async ops (issued last iteration) are complete when
        // ASYNCcnt drops to the count just issued (in-order completion).
        if constexpr (TRANSB) {
            if (hasNext) asm volatile("s_wait_asynccnt 4" ::: "memory");
            else         asm volatile("s_wait_asynccnt 0" ::: "memory");
        } else {
            if (hasNext) asm volatile("s_wait_asynccnt 2" ::: "memory");
            else         asm volatile("s_wait_asynccnt 0" ::: "memory");
        }
        __syncthreads();

        // ---- 16 x v_wmma_f32_16x16x4_f32 per K-slab per wave ----
        #pragma unroll
        for (int kk = 0; kk < 4; ++kk) {
            // A frag: lanes 0-15 K=4kk+{0,1}; lanes 16-31 K=4kk+{2,3}  (one b64)
            v2f a = *(const v2f*)&Alds[buf][aRowOff + (kk << 2)];
            #pragma unroll
            for (int j = 0; j < 4; ++j) {
                v2f b;
                if constexpr (TRANSB) {
                    // BldsT[n][k]: {B[k][n], B[k+1][n]} contiguous  (one b64)
                    b = *(const v2f*)&BldsT[buf][((j << 4) + l16) * R_STRIDE +
                                                 (kk << 2) + (lh << 1)];
                } else {
                    // pair-interleaved: {B[2p][n], B[2p+1][n]}      (one b64)
                    const int p = (kk << 1) + lh;
                    b = *(const v2f*)&Blds[p * B_STRIDE + (((j << 4) + l16) << 1)];
                }
                acc[j] = __builtin_amdgcn_wmma_f32_16x16x4_f32(
                    false, a, false, b, (short)0, acc[j], false, false);
            }
        }
        __syncthreads();
        buf ^= 1;
    }

    // ---- epilogue + store (C layout: VGPR r -> row rowM+r, lane -> col) ----
    for (int j = 0; j < 4; ++j) {
        int n = nBase + (j << 4) + l16;
        for (int r = 0; r < 8; ++r) {
            int m = rowM + r;
            float val = acc[j][r];
            if (CAUSAL)  { val *= scale; if (n > m) val = NEG_BIG; }
            if (BIASRES) { val += bias[n] + resid[(size_t)m * ldres + n]; }
            Ch[(size_t)m * ldc + n] = val;
        }
    }
}

// ---------------------------------------------------------------------------
// In-place row softmax over rows of length L=4096; one 256-thread block/row,
// 16 values per thread held in registers (single read + single write pass).
// ---------------------------------------------------------------------------
__global__ __launch_bounds__(256)
void softmax_rows(float* __restrict__ attn, int L)
{
    __shared__ float red[256];
    float* p = attn + (size_t)blockIdx.x * L;
    const int t = threadIdx.x;

    float v[16];
    float mx = NEG_BIG;
    #pragma unroll
    for (int i = 0; i < 16; ++i) { v[i] = p[t + i * 256]; mx = fmaxf(mx, v[i]); }
    red[t] = mx; __syncthreads();
    for (int s = 128; s > 0; s >>= 1) {
        if (t < s) red[t] = fmaxf(red[t], red[t + s]);
        __syncthreads();
    }
    mx = red[0]; __syncthreads();

    float sum = 0.f;
    #pragma unroll
    for (int i = 0; i < 16; ++i) { v[i] = __expf(v[i] - mx); sum += v[i]; }
    red[t] = sum; __syncthreads();
    for (int s = 128; s > 0; s >>= 1) {
        if (t < s) red[t] += red[t + s];
        __syncthreads();
    }
    const float inv = 1.0f / red[0];
    #pragma unroll
    for (int i = 0; i < 16; ++i) p[t + i * 256] = v[i] * inv;
}

// ---------------------------------------------------------------------------
// LayerNorm over D=512; one 256-thread block per row, 2 elements per thread.
// ---------------------------------------------------------------------------
__global__ __launch_bounds__(256)
void layernorm_rows(const float* __restrict__ x,
                    const float* __restrict__ gamma,
                    const float* __restrict__ beta,
                    float* __restrict__ out, int D)
{
    __shared__ float rs[256];
    __shared__ float rq[256];
    const int t = threadIdx.x;
    const float* p = x + (size_t)blockIdx.x * D;

    float a = p[t], b = p[t + 256];
    rs[t] = a + b;
    rq[t] = a * a + b * b;
    __syncthreads();
    for (int s = 128; s > 0; s >>= 1) {
        if (t < s) { rs[t] += rs[t + s]; rq[t] += rq[t + s]; }
        __syncthreads();
    }
    const float mu   = rs[0] / (float)D;
    const float var  = rq[0] / (float)D - mu * mu;
    const float rstd = rsqrtf(var + 1e-5f);

    float* o = out + (size_t)blockIdx.x * D;
    o[t]       = (a - mu) * rstd * gamma[t]       + beta[t];
    o[t + 256] = (b - mu) * rstd * gamma[t + 256] + beta[t + 256];
}

// ---------------------------------------------------------------------------
extern "C" void kernel_launch(void* const* d_in, const int* in_sizes, int n_in,
                              void* d_out, int out_size, void* d_ws, size_t ws_size,
                              hipStream_t stream)
{
    (void)in_sizes; (void)n_in; (void)out_size; (void)ws_size;

    const int L = 4096, D = 512, H = 8;

    const float* q      = (const float*)d_in[0];
    const float* k      = (const float*)d_in[1];
    const float* v      = (const float*)d_in[2];
    /* d_in[3] = attn_mask (bool causal) -- implied, unused */
    const float* w_qs   = (const float*)d_in[4];
    const float* w_ks   = (const float*)d_in[5];
    const float* w_vs   = (const float*)d_in[6];
    const float* proj_w = (const float*)d_in[7];
    const float* proj_b = (const float*)d_in[8];
    const float* gamma  = (const float*)d_in[9];
    const float* beta   = (const float*)d_in[10];

    float* out  = (float*)d_out;                 // (L, D)
    float* attn = out + (size_t)L * D;           // (H, L, L) — built in place

    const size_t LD = (size_t)L * D;             // 2M floats
    float* ws    = (float*)d_ws;
    float* tmp   = ws;                           // (H, L, D)
    float* qs    = tmp   + 8 * LD;               // (H, L, D)
    float* ks    = qs    + 8 * LD;               // (H, L, D)
    float* vs    = ks    + 8 * LD;               // (H, L, D)
    float* heads = vs    + 8 * LD;               // (L, H*D)
    float* preln = heads + 8 * LD;               // (L, D)

    const dim3 blk(128);

    // 1) q_s/k_s/v_s = (X @ W_h) @ W_h   (batched over heads)
    const dim3 gProj(D / 64, L / 64, H);
    const float* X[3] = { q, k, v };
    const float* W[3] = { w_qs, w_ks, w_vs };
    float*       S[3] = { qs, ks, vs };
    for (int t = 0; t < 3; ++t) {
        gemm_wmma<false, false, false><<<gProj, blk, 0, stream>>>(
            X[t], W[t], tmp, D, D, D, D,
            0L, (long)D * D, (long)LD, 0, 1.0f, nullptr, nullptr, 0);
        gemm_wmma<false, false, false><<<gProj, blk, 0, stream>>>(
            tmp, W[t], S[t], D, D, D, D,
            (long)LD, (long)D * D, (long)LD, 0, 1.0f, nullptr, nullptr, 0);
    }

    // 2) scores = q_s @ k_s^T / sqrt(D), causal mask -> sentinel
    const dim3 gScore(L / 64, L / 64, H);
    gemm_wmma<true, true, false><<<gScore, blk, 0, stream>>>(
        qs, ks, attn, D, D, D, L,
        (long)LD, (long)LD, (long)L * (long)L, 0,
        0.044194173824159216f /* 1/sqrt(512) */, nullptr, nullptr, 0);

    // 3) attn = softmax(scores) in place
    softmax_rows<<<dim3(H * L), dim3(256), 0, stream>>>(attn, L);

    // 4) heads[:, h*D:(h+1)*D] = attn[h] @ v_s[h]
    const dim3 gHeads(D / 64, L / 64, H);
    gemm_wmma<false, false, false><<<gHeads, blk, 0, stream>>>(
        attn, vs, heads, L, L, D, H * D,
        (long)L * (long)L, (long)LD, 0L, D, 1.0f, nullptr, nullptr, 0);

    // 5) preln = heads @ proj_w^T + proj_b + q   (residual fused)
    const dim3 gFinal(D / 64, L / 64, 1);
    gemm_wmma<true, false, true><<<gFinal, blk, 0, stream>>>(
        heads, proj_w, preln, H * D, H * D, H * D, D,
        0L, 0L, 0L, 0, 1.0f, proj_b, q, D);

    // 6) out = LayerNorm(preln)
    layernorm_rows<<<dim3(L), dim3(256), 0, stream>>>(preln, gamma, beta, out, D);
}